// sharedWeightLayer_89584427860409
// MI455X (gfx1250) — compile-verified
//
#include <hip/hip_runtime.h>
#include <hip/hip_bf16.h>
#include <stdint.h>

// Output is a banded broadcast: Diag on d==0, Off[d] on 1<=d<=9,
// Off[-d+10] on -9<=d<=-1, zero elsewhere (x never contributes to the
// selected values). Pure store-bandwidth kernel: 256 MiB of fp32 stores,
// ~11.5 us floor @ 23.3 TB/s. Bulk zeros are streamed by the CDNA5 Tensor
// Data Mover (iterate mode, one 32 KB LDS zero tile reused every iteration);
// the 19-wide diagonal band is overwritten after s_wait_tensorcnt.

#define N_DIM            8192
#define ROWS_PER_BLOCK   32
#define ZQWORDS          4096   // 4096 x 8B = 32 KB LDS zero tile
#define ZQUADS           (ZQWORDS / 2)  // as 16B quads
#define TDM_ITERS        (ROWS_PER_BLOCK * N_DIM / (2 * ZQWORDS))  // 32

typedef unsigned int v4u __attribute__((ext_vector_type(4)));
typedef int          v4i __attribute__((ext_vector_type(4)));
typedef int          v8i __attribute__((ext_vector_type(8)));

__global__ __launch_bounds__(256) void band_broadcast_kernel(
    const float* __restrict__ off,   // 20 floats; indices 1..9 and 11..19 used
    const float* __restrict__ diag,  // 1 float
    float* __restrict__ out)
{
    __shared__ v4u  zbuf[ZQUADS];   // 32 KB of zeros (TDM source)
    __shared__ float lut[19];       // band values, indexed by k = d + 9

    const int tid = (int)threadIdx.x;
    const int r0  = (int)blockIdx.x * ROWS_PER_BLOCK;

    // --- 1) zero the LDS staging tile (ds_store_b128) + stage band LUT ---
    for (int i = tid; i < ZQUADS; i += (int)blockDim.x) {
        v4u z = { 0u, 0u, 0u, 0u };
        zbuf[i] = z;
    }
    if (tid < 19) {
        int d = tid - 9;                       // d in [-9, 9]
        float v;
        if (d == 0)      v = diag[0];
        else if (d > 0)  v = off[d];           // super-diagonals: Off[1..9]
        else             v = off[10 - d];      // sub-diagonals:   Off[11..19]
        lut[tid] = v;
    }
    __syncthreads();

    // --- 2) wave 0 issues one TDM store: 32 iterations x 32KB = 1 MiB of
    //        zeros covering this block's contiguous 32-row slab ---
    if (tid < 32) {  // exactly wave 0 (wave32)
        uint64_t gaddr = (uint64_t)(uintptr_t)out +
                         (uint64_t)r0 * (uint64_t)(N_DIM * 4);
        // low 32 bits of the flat shared pointer == LDS byte offset
        uint32_t lds = (uint32_t)(uintptr_t)&zbuf[0];

        // D# group 0: count=1 (valid user desc), lds_addr, global_addr, type=2
        v4u g0 = { 1u,
                   lds,
                   (uint32_t)gaddr,
                   ((uint32_t)(gaddr >> 32) & 0x01FFFFFFu) | (2u << 30) };

        // D# group 1:
        //  dw0: data_size=3 (8B) <<16 | iterate_enable <<19
        //  dw1: [63:48] = tensor_dim0[15:0] = 4096
        //  dw2: [79:64] = tensor_dim0[31:16] = 0, [95:80] = tensor_dim1[15:0]=1
        //  dw3: [111:96] = tensor_dim1[31:16] = 0, [127:112] = tile_dim0 = 4096
        //  dw4: tile_dim1 = 0 (1D tile), tile_dim2 = 0
        //  dw5: tensor_dim0_stride = 4096 (8B units)
        v8i g1 = { (int)((3u << 16) | (1u << 19)),
                   (int)((unsigned)ZQWORDS << 16),
                   (int)(1u << 16),
                   (int)((unsigned)ZQWORDS << 16),
                   0,
                   (int)ZQWORDS,
                   0,
                   0 };

        // D# group 2 (iterate mode):
        //  dw1 = lds_addr_increment = 0 (reuse same LDS zeros each iteration)
        //  dw2 = global_addr_increment[31:0] = 4096 (8B units -> 32 KB/iter)
        //  dw3: [15:0] ga_inc[47:32]=0, [31:16] iterate_count = TDM_ITERS-1
        v4i g2 = { 0, 0, (int)ZQWORDS, (int)((unsigned)(TDM_ITERS - 1) << 16) };
        v4i g3 = { 0, 0, 0, 0 };

#if __clang_major__ >= 23
        v8i g4 = { 0, 0, 0, 0, 0, 0, 0, 0 };
        __builtin_amdgcn_tensor_store_from_lds(g0, g1, g2, g3, g4, 0);
#else
        __builtin_amdgcn_tensor_store_from_lds(g0, g1, g2, g3, 0);
#endif
        __builtin_amdgcn_s_wait_tensorcnt(0);
    }
    __syncthreads();  // zeros committed before the band overwrite

    // --- 3) overwrite the 19-wide diagonal band for our 32 rows ---
    for (int u = tid; u < ROWS_PER_BLOCK * 19; u += (int)blockDim.x) {
        int row = r0 + u / 19;
        int k   = u % 19;
        int col = row + (k - 9);
        if ((unsigned)col < (unsigned)N_DIM) {
            out[(size_t)row * N_DIM + col] = lut[k];
        }
    }
}

extern "C" void kernel_launch(void* const* d_in, const int* in_sizes, int n_in,
                              void* d_out, int out_size, void* d_ws, size_t ws_size,
                              hipStream_t stream) {
    (void)in_sizes; (void)n_in; (void)d_ws; (void)ws_size; (void)out_size;
    // d_in[0] = x (8192x8192 f32) -- mathematically unused by the reference
    const float* off  = (const float*)d_in[1];  // 20 floats
    const float* diag = (const float*)d_in[2];  // 1 float
    float* out = (float*)d_out;

    dim3 grid(N_DIM / ROWS_PER_BLOCK);  // 256 blocks
    dim3 block(256);                    // 8 waves (wave32)
    band_broadcast_kernel<<<grid, block, 0, stream>>>(off, diag, out);
}